// QOuter_2860448219733
// MI455X (gfx1250) — compile-verified
//
#include <hip/hip_runtime.h>

// QOuter: out_rr[b] = r r^T + i i^T ; out_ii[b] = i r^T - r i^T   (B=64, N=1024, f32)
// Store-bandwidth bound (536 MB out => ~23 us at 23.3 TB/s).
// Each wave32 computes one 16x16 tile of both outputs via V_WMMA_F32_16X16X4_F32
// (K=2 rank-2 update padded to K=4 with zero columns), then streams the tiles
// out with non-temporal b32 stores (64B contiguous per half-wave per register).

typedef __attribute__((ext_vector_type(2))) float v2f;
typedef __attribute__((ext_vector_type(8))) float v8f;

#define BATCHES 64
#define NDIM 1024
#define TILES_PER_DIM 64            // 1024 / 16
#define WAVES_PER_BLOCK 8           // 256 threads
#define TOTAL_TILES (BATCHES * TILES_PER_DIM * TILES_PER_DIM)  // 262144

__global__ __launch_bounds__(256) void qouter_wmma_kernel(
    const float* __restrict__ real,
    const float* __restrict__ imag,
    float* __restrict__ out)
{
    const int lane = threadIdx.x & 31;
    const int wave = threadIdx.x >> 5;

    const unsigned tile = blockIdx.x * WAVES_PER_BLOCK + wave;
    const int tj = tile & (TILES_PER_DIM - 1);          // j-tile
    const int ti = (tile >> 6) & (TILES_PER_DIM - 1);   // i-tile
    const int b  = tile >> 12;                          // batch

    const int i0  = ti << 4;
    const int j0  = tj << 4;
    const int l15 = lane & 15;
    const bool lo = lane < 16;   // lanes 0-15 carry K=0,1; lanes 16-31 carry K=2,3 (zero pad)

    const float* __restrict__ rb = real + ((size_t)b << 10);
    const float* __restrict__ ib = imag + ((size_t)b << 10);

    // Row values for the A (i-index) side and B (j-index) side of the tile.
    const float rA = rb[i0 + l15];
    const float iA = ib[i0 + l15];
    const float rB = rb[j0 + l15];
    const float iB = ib[j0 + l15];

    // A-matrix 16x4 layout: lanes 0-15 -> {K0,K1} in VGPR0/1; lanes 16-31 -> {K2,K3} = 0.
    v2f a_rr, a_ii, bmat;
    a_rr.x = lo ? rA : 0.0f;   a_rr.y = lo ? iA  : 0.0f;   // out_rr row: [r_m, i_m, 0, 0]
    a_ii.x = lo ? iA : 0.0f;   a_ii.y = lo ? -rA : 0.0f;   // out_ii row: [i_m, -r_m, 0, 0]
    // B-matrix 4x16 layout (mirrored): lanes 0-15 -> rows K0={r_n}, K1={i_n}; lanes 16-31 -> 0.
    bmat.x = lo ? rB : 0.0f;   bmat.y = lo ? iB  : 0.0f;

    const v8f zero = {};
    // D = A x B + 0 : 16x16 f32 tile, 8 VGPRs.
    v8f d_rr = __builtin_amdgcn_wmma_f32_16x16x4_f32(
        false, a_rr, false, bmat, (short)0, zero, false, false);
    v8f d_ii = __builtin_amdgcn_wmma_f32_16x16x4_f32(
        false, a_ii, false, bmat, (short)0, zero, false, false);

    // D layout: VGPR v, lane l -> row M = v + 8*(l>=16), col N = l&15.
    const int hi8 = (lane >> 4) << 3;   // 0 for lanes 0-15, 8 for lanes 16-31
    float* __restrict__ out_rr = out;
    float* __restrict__ out_ii = out + ((size_t)BATCHES << 20);  // 64*1024*1024 floats

    size_t base = ((size_t)b << 20) + ((size_t)(i0 + hi8) << 10) + (size_t)(j0 + l15);
#pragma unroll
    for (int v = 0; v < 8; ++v) {
        const size_t idx = base + ((size_t)v << 10);   // next row of the tile
        __builtin_nontemporal_store(d_rr[v], &out_rr[idx]);
        __builtin_nontemporal_store(d_ii[v], &out_ii[idx]);
    }
}

extern "C" void kernel_launch(void* const* d_in, const int* in_sizes, int n_in,
                              void* d_out, int out_size, void* d_ws, size_t ws_size,
                              hipStream_t stream) {
    (void)in_sizes; (void)n_in; (void)out_size; (void)d_ws; (void)ws_size;
    const float* real = (const float*)d_in[0];
    const float* imag = (const float*)d_in[1];
    float* out = (float*)d_out;

    const int blocks = TOTAL_TILES / WAVES_PER_BLOCK;   // 32768
    qouter_wmma_kernel<<<blocks, 256, 0, stream>>>(real, imag, out);
}